// NCCLoss_71451075936663
// MI455X (gfx1250) — compile-verified
//
#include <hip/hip_runtime.h>

// ---------------------------------------------------------------------------
// Local NCC loss, window 9^3, volumes (2,1,160,192,160) fp32 -> scalar.
// Pass 1: W-axis box sums of 5 channels (I, J, I2, J2, IJ) -> ws buffer A.
// Pass 2: H-axis box sums via V_WMMA_F32_16X16X4_F32 (banded-ones matmul),
//         fused D-axis running box sum (LDS ring of 9 slices), NCC math,
//         wave reduction, atomic accumulate.  Branch-free halo loads
//         (clamped address + 0/1 mask) so all 60 loads per slice batch
//         ahead of the 30 WMMAs.
// Pass 3: finalize scalar  out = -sum(cc)/N.
// ---------------------------------------------------------------------------

typedef float v2f __attribute__((ext_vector_type(2)));
typedef float v8f __attribute__((ext_vector_type(8)));

#define BB   2
#define DD   160
#define HH   192
#define WW   160
#define WIN  9
#define PAD  4
#define ZC   16            // z-chunk per block in pass 2 (DD % ZC == 0)
#define NTOT ((size_t)BB * DD * HH * WW)   // 9,830,400

// ---------------- Pass 1: sliding 9-tap sums along W (contiguous axis) ------
__global__ __launch_bounds__(320) void ncc_pass_w(const float* __restrict__ Jp,  // y_pred
                                                  const float* __restrict__ Ip,  // y_true
                                                  float* __restrict__ A) {
    __shared__ float LI[2][WW + 2 * PAD];
    __shared__ float LJ[2][WW + 2 * PAD];

    const int x  = threadIdx.x;           // 0..159
    const int ty = threadIdx.y;           // 0..1
    const int y  = blockIdx.x * 2 + ty;   // 0..191
    const int z  = blockIdx.y;            // 0..159
    const int b  = blockIdx.z;            // 0..1

    const size_t idx = (((size_t)(b * DD + z) * HH + y) * WW + x);

    const float iv = Ip[idx];
    const float jv = Jp[idx];
    LI[ty][x + PAD] = iv;
    LJ[ty][x + PAD] = jv;
    if (x < PAD) {                        // zero halo both sides
        LI[ty][x] = 0.0f;            LJ[ty][x] = 0.0f;
        LI[ty][WW + PAD + x] = 0.0f; LJ[ty][WW + PAD + x] = 0.0f;
    }
    __syncthreads();

    float si = 0.f, sj = 0.f, sii = 0.f, sjj = 0.f, sij = 0.f;
#pragma unroll
    for (int t = 0; t < WIN; ++t) {
        const float a = LI[ty][x + t];
        const float c = LJ[ty][x + t];
        si  += a;      sj  += c;
        sii += a * a;  sjj += c * c;  sij += a * c;
    }
    A[0 * NTOT + idx] = si;
    A[1 * NTOT + idx] = sj;
    A[2 * NTOT + idx] = sii;
    A[3 * NTOT + idx] = sjj;
    A[4 * NTOT + idx] = sij;
}

// ------- Pass 2: WMMA H-sums, fused running D-sum + NCC + reduction ---------
__global__ __launch_bounds__(32) void ncc_pass_hd(const float* __restrict__ A,
                                                  double* __restrict__ accum) {
    // ring[slice][channel][lane][elem] : 9 * 5 * 32 * 8 * 4B = 46,080 B
    __shared__ float ring[9][5][32][8];

    const int lane = threadIdx.x;                 // wave32
    const int x0   = blockIdx.x * 16;             // 10 x-tiles
    const int wy0  = blockIdx.y * 16;             // 12 y-tiles
    const int b    = blockIdx.z / (DD / ZC);
    const int z0   = (blockIdx.z % (DD / ZC)) * ZC;

    // A-operand: banded ones, 16x24 split into six 16x4 chunks.
    // f32 A layout: lanes 0-15 -> M, VGPR0 holds {K0 | K2}, VGPR1 {K1 | K3}.
    const int m   = lane & 15;
    const int kl0 = (lane < 16) ? 0 : 2;          // K slot of component 0
    v2f band[6];
#pragma unroll
    for (int kb = 0; kb < 6; ++kb) {
        const int k0 = 4 * kb + kl0;
        const int k1 = k0 + 1;
        band[kb][0] = (k0 >= m && k0 <= m + 8) ? 1.0f : 0.0f;
        band[kb][1] = (k1 >= m && k1 <= m + 8) ? 1.0f : 0.0f;
    }
    const int col = x0 + (lane & 15);             // B column owned by this lane

    // Branch-free halo: clamped in-slice element offsets + 0/1 masks,
    // computed once (validity depends only on y, not z).
    int   off0[6], off1[6];
    float msk0[6], msk1[6];
#pragma unroll
    for (int kb = 0; kb < 6; ++kb) {
        const int row0 = wy0 - 4 + 4 * kb + kl0;
        const int row1 = row0 + 1;
        const int cr0  = row0 < 0 ? 0 : (row0 >= HH ? HH - 1 : row0);
        const int cr1  = row1 < 0 ? 0 : (row1 >= HH ? HH - 1 : row1);
        off0[kb] = cr0 * WW + col;
        off1[kb] = cr1 * WW + col;
        msk0[kb] = (row0 >= 0 && row0 < HH) ? 1.0f : 0.0f;
        msk1[kb] = (row1 >= 0 && row1 < HH) ? 1.0f : 0.0f;
    }

    float acc[5][8];
#pragma unroll
    for (int c = 0; c < 5; ++c)
#pragma unroll
        for (int i = 0; i < 8; ++i) acc[c][i] = 0.0f;

    double local = 0.0;
    const float inv_n = 1.0f / 729.0f;

    for (int s = 0; s < ZC + 8; ++s) {            // z-sweep with 8-slice warmup
        const int zin = z0 - 4 + s;
        const int r   = s % 9;

        float newt[5][8];
        if (zin >= 0 && zin < DD) {               // wave-uniform branch
            const size_t slice = (size_t)(b * DD + zin) * HH * WW;

            // 1) Batch all 60 unconditional loads for this slice.
            float bvals[5][6][2];
#pragma unroll
            for (int c = 0; c < 5; ++c) {
                const float* __restrict__ sbase = A + (size_t)c * NTOT + slice;
#pragma unroll
                for (int kb = 0; kb < 6; ++kb) {
                    bvals[c][kb][0] = sbase[off0[kb]];
                    bvals[c][kb][1] = sbase[off1[kb]];
                }
            }
            // 2) Apply halo masks (VALU, overlaps load drain).
#pragma unroll
            for (int c = 0; c < 5; ++c)
#pragma unroll
                for (int kb = 0; kb < 6; ++kb) {
                    bvals[c][kb][0] *= msk0[kb];
                    bvals[c][kb][1] *= msk1[kb];
                }
            // 3) 30 WMMAs: kb outer / channel inner -> 5 independent chains.
            v8f d[5];
#pragma unroll
            for (int c = 0; c < 5; ++c) d[c] = (v8f){0.f,0.f,0.f,0.f,0.f,0.f,0.f,0.f};
#pragma unroll
            for (int kb = 0; kb < 6; ++kb)
#pragma unroll
                for (int c = 0; c < 5; ++c) {
                    v2f bv;
                    bv[0] = bvals[c][kb][0];
                    bv[1] = bvals[c][kb][1];
                    d[c] = __builtin_amdgcn_wmma_f32_16x16x4_f32(
                               /*neg_a=*/false, band[kb],
                               /*neg_b=*/false, bv,
                               /*c_mod=*/(short)0, d[c],
                               /*reuse_a=*/false, /*reuse_b=*/false);
                }
#pragma unroll
            for (int c = 0; c < 5; ++c)
#pragma unroll
                for (int i = 0; i < 8; ++i) newt[c][i] = d[c][i];
        } else {
#pragma unroll
            for (int c = 0; c < 5; ++c)
#pragma unroll
                for (int i = 0; i < 8; ++i) newt[c][i] = 0.0f;
        }

        // running D-axis sum: acc += new - slice_leaving_window
#pragma unroll
        for (int c = 0; c < 5; ++c)
#pragma unroll
            for (int i = 0; i < 8; ++i) {
                const float old = (s >= 9) ? ring[r][c][lane][i] : 0.0f;
                acc[c][i] += newt[c][i] - old;
                ring[r][c][lane][i] = newt[c][i];
            }

        if (s >= 8) {                             // output slice zout = zin - 4
#pragma unroll
            for (int i = 0; i < 8; ++i) {
                const float Si  = acc[0][i];
                const float Sj  = acc[1][i];
                const float Sii = acc[2][i];
                const float Sjj = acc[3][i];
                const float Sij = acc[4][i];
                const float cross = Sij - Si * Sj * inv_n;
                const float ivar  = Sii - Si * Si * inv_n;
                const float jvar  = Sjj - Sj * Sj * inv_n;
                const float cc    = cross * cross / (ivar * jvar + 1e-5f);
                local += (double)cc;
            }
        }
    }

    // wave32 reduction, then one atomic per block
#pragma unroll
    for (int off = 16; off > 0; off >>= 1)
        local += __shfl_down(local, off, 32);
    if (lane == 0) atomicAdd(accum, local);
}

// ---------------- Pass 3: finalize scalar -----------------------------------
__global__ void ncc_finalize(const double* __restrict__ accum,
                             float* __restrict__ out) {
    out[0] = (float)(-accum[0] / (double)NTOT);
}

// ---------------------------------------------------------------------------
extern "C" void kernel_launch(void* const* d_in, const int* in_sizes, int n_in,
                              void* d_out, int out_size, void* d_ws, size_t ws_size,
                              hipStream_t stream) {
    const float* Jp = (const float*)d_in[0];   // y_pred
    const float* Ip = (const float*)d_in[1];   // y_true
    float*  A     = (float*)d_ws;                               // 5*N floats
    double* accum = (double*)((char*)d_ws + 5 * NTOT * sizeof(float));

    hipMemsetAsync(accum, 0, sizeof(double), stream);

    dim3 b1(WW, 2, 1), g1(HH / 2, DD, BB);
    ncc_pass_w<<<g1, b1, 0, stream>>>(Jp, Ip, A);

    dim3 b2(32, 1, 1), g2(WW / 16, HH / 16, BB * (DD / ZC));
    ncc_pass_hd<<<g2, b2, 0, stream>>>(A, accum);

    ncc_finalize<<<1, 1, 0, stream>>>(accum, (float*)d_out);
}